// JensenShannonDivergence_68504728371333
// MI455X (gfx1250) — compile-verified
//
#include <hip/hip_runtime.h>

// ---------------------------------------------------------------------------
// KDE + Jensen-Shannon divergence for MI455X (gfx1250, wave32).
//
// Hot loop: V_WMMA_F32_16X16X4_F32 computes -c*d^2 for a 16x16 (grid x data)
// tile in ONE instruction using augmented coordinates:
//   A[i,:] = [2c*gx, 2c*gy, 1, -c*(gx^2+gy^2)]
//   B[:,j] = [xj, yj, -c*(xj^2+yj^2), 1]^T
//   (A*B)[i,j] = -c * ||g_i - x_j||^2,   c = log2(e) / (2 h^2)
// followed by v_exp_f32 (exp2) per element. One transcendental per pair is
// the irreducible cost; WMMA removes all distance VALU work. The data load
// for chunk n+1 is issued before the WMMA/exp block of chunk n (software
// pipelining) so no load latency sits on the critical path.
// ---------------------------------------------------------------------------

typedef __attribute__((ext_vector_type(2))) float v2f;
typedef __attribute__((ext_vector_type(8))) float v8f;

#define NSLICE 8
#define H_BW   0.05f

__global__ __launch_bounds__(256) void kde_tile_kernel(
    const float* __restrict__ data, const float* __restrict__ grid,
    float* __restrict__ partial, int N, int G)
{
    const int lane = threadIdx.x & 31;
    const int wave = (int)((blockIdx.x * blockDim.x + threadIdx.x) >> 5);
    const int numTilesM  = (G + 15) >> 4;
    const int totalWaves = numTilesM * NSLICE;
    if (wave >= totalWaves) return;              // whole-wave exit only

    const int tileM = wave % numTilesM;
    const int slice = wave / numTilesM;

    const float c2 = (1.0f / (2.0f * H_BW * H_BW)) * 1.44269504088896340736f;
    const int  r  = lane & 15;
    const bool hi = lane >= 16;

    // Loop-invariant A fragment: 16 grid rows of this tile.
    int gi = tileM * 16 + r; if (gi >= G) gi = G - 1;
    const float gx = grid[2 * gi];
    const float gy = grid[2 * gi + 1];
    v2f afrag;
    afrag.x = hi ? 1.0f                         : (2.0f * c2 * gx);   // K=2 | K=0
    afrag.y = hi ? (-c2 * (gx * gx + gy * gy))  : (2.0f * c2 * gy);   // K=3 | K=1

    // N-slice bounds (16-aligned chunk starts).
    const int chunk = ((((N + NSLICE - 1) / NSLICE) + 15) >> 4) << 4;
    const int n0 = slice * chunk;
    int n1 = n0 + chunk; if (n1 > N) n1 = N;

    v8f acc = {0.f, 0.f, 0.f, 0.f, 0.f, 0.f, 0.f, 0.f};
    const v8f czero = {0.f, 0.f, 0.f, 0.f, 0.f, 0.f, 0.f, 0.f};

    // Clamped chunk load: always a legal address, pad lanes get huge coords
    // so their column evaluates to exp2(-inf) = 0.
    auto load_chunk = [&](int nb, float& xv, float& yv) {
        int j  = nb + r;
        int jc = (j < N) ? j : (N - 1);
        if (jc < 0) jc = 0;
        xv = data[2 * jc];
        yv = data[2 * jc + 1];
        if (j >= N) { xv = 1.0e30f; yv = 1.0e30f; }
    };

    float nx = 1.0e30f, ny = 1.0e30f;
    if (n0 < n1) load_chunk(n0, nx, ny);

    for (int nb = n0; nb < n1; nb += 16) {
        const float cx = nx, cy = ny;            // consumes PREVIOUS load
        load_chunk(nb + 16, nx, ny);             // issue NEXT load now
        // Prefetch farther ahead for the first (HBM) pass.
        {
            int pj = nb + 512 + r; if (pj >= N) pj = N - 1;
            __builtin_prefetch(&data[2 * pj], 0, 1);
        }

        v2f bfrag;
        bfrag.x = hi ? (-c2 * (cx * cx + cy * cy)) : cx;   // K=2 | K=0
        bfrag.y = hi ? 1.0f                        : cy;   // K=3 | K=1

        // D[i,j] = -c * d^2 for the whole 16x16 tile (all K=4 slots useful).
        v8f d = __builtin_amdgcn_wmma_f32_16x16x4_f32(
            false, afrag, false, bfrag, (short)0, czero, false, false);

#pragma unroll
        for (int v = 0; v < 8; ++v)
            acc[v] += __builtin_amdgcn_exp2f(d[v]);        // v_exp_f32
    }

    // Column (N) reduction: 16-lane xor tree within each half-wave.
#pragma unroll
    for (int v = 0; v < 8; ++v) {
#pragma unroll
        for (int off = 8; off >= 1; off >>= 1)
            acc[v] += __shfl_xor(acc[v], off, 16);
    }

    if ((lane & 15) == 0) {
        const int rowBase = tileM * 16 + (hi ? 8 : 0);
#pragma unroll
        for (int v = 0; v < 8; ++v) {
            const int row = rowBase + v;
            if (row < G) partial[(size_t)slice * G + row] = acc[v];
        }
    }
}

// ---- epilogue: combine slices, normalize, JSD (tiny vs. the KDE) ----------

__device__ __forceinline__ float block_reduce_256(float v, float* red)
{
    red[threadIdx.x] = v;
    __syncthreads();
    for (int st = 128; st > 0; st >>= 1) {
        if ((int)threadIdx.x < st) red[threadIdx.x] += red[threadIdx.x + st];
        __syncthreads();
    }
    return red[0];
}

__global__ __launch_bounds__(256) void combine_kernel(
    const float* __restrict__ partial, float* __restrict__ praw,
    float* __restrict__ bsums, int G)
{
    __shared__ float red[256];
    const int i = blockIdx.x * 256 + threadIdx.x;
    float v = 0.f;
    if (i < G) {
#pragma unroll
        for (int s = 0; s < NSLICE; ++s) v += partial[(size_t)s * G + i];
        praw[i] = v;
    }
    const float t = block_reduce_256(v, red);
    if (threadIdx.x == 0) bsums[blockIdx.x] = t;
}

__global__ __launch_bounds__(256) void reduce_kernel(
    const float* __restrict__ in, int n, float* __restrict__ out)
{
    __shared__ float red[256];
    float v = 0.f;
    for (int i = threadIdx.x; i < n; i += 256) v += in[i];
    const float t = block_reduce_256(v, red);
    if (threadIdx.x == 0) out[0] = t;
}

__global__ __launch_bounds__(256) void jsd_kernel(
    const float* __restrict__ praw, const float* __restrict__ q,
    const float* __restrict__ Sptr, float* __restrict__ bsums, int G)
{
    __shared__ float red[256];
    const int i = blockIdx.x * 256 + threadIdx.x;
    float t = 0.f;
    if (i < G) {
        const float invS = 1.0f / Sptr[0];
        const float p  = praw[i] * invS;
        const float qq = q[i];
        const float m  = 0.5f * (p + qq);
        if (p  > 0.f) t += p  * logf(p  / m);
        if (qq > 0.f) t += qq * logf(qq / m);
        t *= 0.5f;
    }
    const float s = block_reduce_256(t, red);
    if (threadIdx.x == 0) bsums[blockIdx.x] = s;
}

// ---------------------------------------------------------------------------

extern "C" void kernel_launch(void* const* d_in, const int* in_sizes, int n_in,
                              void* d_out, int out_size, void* d_ws, size_t ws_size,
                              hipStream_t stream)
{
    (void)n_in; (void)out_size; (void)ws_size;

    const float* data   = (const float*)d_in[0];   // [N,2]
    const float* grid   = (const float*)d_in[1];   // [G,2]
    const float* target = (const float*)d_in[2];   // [G]
    const int N = in_sizes[0] / 2;
    const int G = in_sizes[1] / 2;

    float* ws = (float*)d_ws;
    float* partial = ws;                                   // NSLICE * G
    float* praw    = partial + (size_t)NSLICE * G;         // G
    const int nblocksG = (G + 255) / 256;
    float* bsumsA  = praw + G;                             // nblocksG
    float* S       = bsumsA + nblocksG;                    // 1
    float* bsumsB  = S + 1;                                // nblocksG
    float* out     = (float*)d_out;

    const int numTilesM  = (G + 15) / 16;
    const int totalWaves = numTilesM * NSLICE;
    const int blocks     = (totalWaves + 7) / 8;           // 8 waves / block

    kde_tile_kernel<<<blocks, 256, 0, stream>>>(data, grid, partial, N, G);
    combine_kernel <<<nblocksG, 256, 0, stream>>>(partial, praw, bsumsA, G);
    reduce_kernel  <<<1, 256, 0, stream>>>(bsumsA, nblocksG, S);
    jsd_kernel     <<<nblocksG, 256, 0, stream>>>(praw, target, S, bsumsB, G);
    reduce_kernel  <<<1, 256, 0, stream>>>(bsumsB, nblocksG, out);
}